// ClassBalancedSupConLoss_55379308314897
// MI455X (gfx1250) — compile-verified
//
#include <hip/hip_runtime.h>
#include <hip/hip_bf16.h>

typedef float v2f __attribute__((ext_vector_type(2)));
typedef float v4f __attribute__((ext_vector_type(4)));
typedef float v8f __attribute__((ext_vector_type(8)));
typedef unsigned int v4u __attribute__((ext_vector_type(4)));
typedef int v8i __attribute__((ext_vector_type(8)));
typedef int v4i __attribute__((ext_vector_type(4)));

#define BATCH_N    2048
#define BANK_M     16384
#define DIM_K      128
#define TOTAL_COLS (BATCH_N + BANK_M)    // 18432
#define COL_CHUNK  512                   // columns per blockIdx.y chunk
#define NCHUNK     36                    // 36 * 512 = 18432
#define TILES_PER_CHUNK (COL_CHUNK / 16) // 32
#define ROWS_PER_WG 128                  // 4 waves * 32 rows
#define BASE_TEMP  0.07f

// LDS B tile: 16 rows of 128 f32, padded to 130 dwords/row to avoid bank
// conflicts (matches TDM pad_interval=128 dwords, pad_amount=2 dwords).
#define LDS_ROW_DW 130
#define LDS_TILE_DW (16 * LDS_ROW_DW)

#if defined(__has_builtin)
#if __has_builtin(__builtin_amdgcn_tensor_load_to_lds)
#define USE_TDM 1
#endif
#endif

// ---------------------------------------------------------------------------
__global__ void zero_ws_kernel(float* __restrict__ w, int n) {
    int i = blockIdx.x * blockDim.x + threadIdx.x;
    if (i < n) w[i] = 0.0f;
}

// ---------------------------------------------------------------------------
// Fused SupCon kernel.
// 4 waves / WG. Each wave holds two 16x128 f32 A tiles in registers (32
// anchor rows). B tiles (16 cols x 128 K) are staged into double-buffered
// LDS by the Tensor Data Mover (TENSOR_LOAD_TO_LDS, TENSORcnt), and each
// ds_load_b64 fragment feeds two V_WMMA_F32_16X16X4_F32 ops.
// ---------------------------------------------------------------------------
__global__ __launch_bounds__(128) void supcon_tile_kernel(
    const float* __restrict__ emb,        // [2048,128]
    const int*   __restrict__ labels,     // [2048]
    const float* __restrict__ bank,       // [16384,128]
    const int*   __restrict__ bank_labels,// [16384]
    const float* __restrict__ temps,      // [3]
    float* __restrict__ denom_acc,        // [2048]
    float* __restrict__ pos_acc)          // [2048]
{
    __shared__ float btile[2][LDS_TILE_DW];

    const int tid   = threadIdx.x;        // 0..127
    const int lane  = tid & 31;
    const int wave  = tid >> 5;           // 0..3
    const int half  = lane >> 4;          // 0: lanes 0-15, 1: lanes 16-31
    const int l16   = lane & 15;
    const int koff  = half * 2;           // K offset per lane half (f32 layout)

    const int row_base  = blockIdx.x * ROWS_PER_WG + wave * 32;
    const int chunk     = blockIdx.y;
    const int col_start = chunk * COL_CHUNK;

    // ---- Preload two A tiles into registers (f32 16x4 A layout) ----------
    const float* arow0 = emb + (size_t)(row_base + l16) * DIM_K;
    const float* arow1 = emb + (size_t)(row_base + 16 + l16) * DIM_K;
    v2f a0[32], a1[32];
#pragma unroll
    for (int ks = 0; ks < 32; ++ks) {
        a0[ks] = *(const v2f*)(arow0 + ks * 4 + koff);
        a1[ks] = *(const v2f*)(arow1 + ks * 4 + koff);
    }

    // ---- Per-lane epilogue metadata --------------------------------------
    float invt0[8], invt1[8];
    int   rlab0[8], rlab1[8];
#pragma unroll
    for (int j = 0; j < 8; ++j) {
        int m0 = row_base + half * 8 + j;
        int m1 = m0 + 16;
        int lb0 = labels[m0];
        int lb1 = labels[m1];
        rlab0[j] = lb0;  invt0[j] = 1.0f / temps[lb0];
        rlab1[j] = lb1;  invt1[j] = 1.0f / temps[lb1];
    }

    float denom0[8], pos0[8], denom1[8], pos1[8];
#pragma unroll
    for (int j = 0; j < 8; ++j) {
        denom0[j] = 0.0f; pos0[j] = 0.0f;
        denom1[j] = 0.0f; pos1[j] = 0.0f;
    }

    // ---- B tile fill: TDM descriptor path (or cooperative fallback) ------
    auto issue_fill = [&](int buf, int ct) {
        const int col0 = col_start + ct * 16;
        const float* src = (col0 < BATCH_N)
                             ? (emb  + (size_t)col0 * DIM_K)
                             : (bank + (size_t)(col0 - BATCH_N) * DIM_K);
#ifdef USE_TDM
        if (tid < 32) {
            const unsigned lds_off = (unsigned)(uintptr_t)(&btile[buf][0]);
            const unsigned long long ga = (unsigned long long)(uintptr_t)src;
            // D# group 0: count=1 | lds_addr | global_addr | type=2
            v4u g0;
            g0.x = 1u;                                   // count=1, no gather
            g0.y = lds_off;                              // lds_addr (bytes)
            g0.z = (unsigned)(ga & 0xffffffffu);         // global_addr[31:0]
            g0.w = (unsigned)((ga >> 32) & 0x01ffffffu)  // global_addr[56:32]
                   | (2u << 30);                         // type = 2 (image)
            // D# group 1: data_size=4B, pad 2 dwords every 128 dwords,
            // tensor 128x16 f32, tile 128x16, dim0 stride 128.
            v8i g1;
            g1[0] = (int)((2u << 16)        // data_size = 4 bytes
                        | (1u << 20)        // pad_enable
                        | (6u << 22)        // pad_interval: 128 dwords
                        | (1u << 25));      // pad_amount: 2 dwords
            g1[1] = (int)(128u << 16);      // tensor_dim0[15:0] = 128
            g1[2] = (int)(16u << 16);       // tensor_dim0 hi=0 | tensor_dim1 lo=16
            g1[3] = (int)(128u << 16);      // tensor_dim1 hi=0 | tile_dim0=128
            g1[4] = (int)16u;               // tile_dim1=16, tile_dim2=0
            g1[5] = (int)128u;              // tensor_dim0_stride[31:0] = 128
            g1[6] = 0;                      // stride hi | tensor_dim1_stride lo
            g1[7] = 0;                      // tensor_dim1_stride hi
            v4i z4 = {0, 0, 0, 0};
            v8i z8 = {0, 0, 0, 0, 0, 0, 0, 0};
            // 6-arg toolchain form: (g0, g1, g2, g3, g4, cpol)
            __builtin_amdgcn_tensor_load_to_lds(g0, g1, z4, z4, z8, 0);
        }
#else
        // Cooperative fill: 8 threads per row, 64B each, honoring the pad.
        {
            const int r = tid >> 3;         // 0..15
            const int s = tid & 7;          // 0..7
            const float* gsrc = src + (size_t)r * DIM_K + s * 16;
            float* dst = &btile[buf][r * LDS_ROW_DW + s * 16];
            v4f t0 = *(const v4f*)(gsrc + 0);
            v4f t1 = *(const v4f*)(gsrc + 4);
            v4f t2 = *(const v4f*)(gsrc + 8);
            v4f t3 = *(const v4f*)(gsrc + 12);
            *(v4f*)(dst + 0)  = t0;
            *(v4f*)(dst + 4)  = t1;
            *(v4f*)(dst + 8)  = t2;
            *(v4f*)(dst + 12) = t3;
        }
#endif
    };

    auto wait_fill = [&]() {
#ifdef USE_TDM
        if (tid < 32) __builtin_amdgcn_s_wait_tensorcnt(0);
#endif
        __syncthreads();
    };

    // ---- Prologue: stage tile 0 ------------------------------------------
    issue_fill(0, 0);
    wait_fill();

    for (int ct = 0; ct < TILES_PER_CHUNK; ++ct) {
        // Kick off DMA of the next tile into the other LDS buffer.
        if (ct + 1 < TILES_PER_CHUNK) issue_fill((ct + 1) & 1, ct + 1);

        const int col0 = col_start + ct * 16;
        const int n    = col0 + l16;          // this lane's column (== its N)

        const bool isbatch = (n < BATCH_N);
        const int  clab    = isbatch ? labels[n] : bank_labels[n - BATCH_N];

        // ---- Two 16x16 f32 tiles per B fragment from LDS -----------------
        const float* bl = &btile[ct & 1][l16 * LDS_ROW_DW + koff];
        v8f c0 = {};
        v8f c1 = {};
#pragma unroll
        for (int ks = 0; ks < 32; ++ks) {
            v2f b = *(const v2f*)(bl + ks * 4);   // ds_load_b64
            c0 = __builtin_amdgcn_wmma_f32_16x16x4_f32(
                     false, a0[ks], false, b, (short)0, c0, false, false);
            c1 = __builtin_amdgcn_wmma_f32_16x16x4_f32(
                     false, a1[ks], false, b, (short)0, c1, false, false);
        }

        // ---- Epilogue: scale, mask, accumulate ---------------------------
#pragma unroll
        for (int j = 0; j < 8; ++j) {
            const int m0 = row_base + half * 8 + j;
            const int m1 = m0 + 16;

            const float logit0 = c0[j] * invt0[j];
            const bool  self0  = isbatch && (n == m0);
            const bool  den0   = isbatch ? (!self0) : (clab != rlab0[j]);
            if (den0) denom0[j] += __expf(logit0);
            if (isbatch && (clab == rlab0[j]) && !self0) pos0[j] += logit0;

            const float logit1 = c1[j] * invt1[j];
            const bool  self1  = isbatch && (n == m1);
            const bool  den1   = isbatch ? (!self1) : (clab != rlab1[j]);
            if (den1) denom1[j] += __expf(logit1);
            if (isbatch && (clab == rlab1[j]) && !self1) pos1[j] += logit1;
        }

        wait_fill();   // DMA of next tile done + all waves finished this tile
    }

    // ---- Reduce over the 16 lanes of each half, then atomically commit ---
#pragma unroll
    for (int j = 0; j < 8; ++j) {
        float d0 = denom0[j], p0 = pos0[j];
        float d1 = denom1[j], p1 = pos1[j];
#pragma unroll
        for (int off = 8; off >= 1; off >>= 1) {
            d0 += __shfl_xor(d0, off, 32);
            p0 += __shfl_xor(p0, off, 32);
            d1 += __shfl_xor(d1, off, 32);
            p1 += __shfl_xor(p1, off, 32);
        }
        if (l16 == 0) {
            const int m0 = row_base + half * 8 + j;
            atomicAdd(&denom_acc[m0],      d0);
            atomicAdd(&pos_acc[m0],        p0);
            atomicAdd(&denom_acc[m0 + 16], d1);
            atomicAdd(&pos_acc[m0 + 16],   p1);
        }
    }
}

// ---------------------------------------------------------------------------
// Final scalar reduction.
// ---------------------------------------------------------------------------
__global__ __launch_bounds__(256) void supcon_finalize_kernel(
    const int*   __restrict__ labels,
    const float* __restrict__ temps,
    const float* __restrict__ denom_acc,
    const float* __restrict__ pos_acc,
    float*       __restrict__ out)
{
    __shared__ int   class_cnt[3];
    __shared__ float red_loss[256];
    __shared__ int   red_valid[256];

    const int t = threadIdx.x;
    if (t < 3) class_cnt[t] = 0;
    __syncthreads();

    for (int i = t; i < BATCH_N; i += 256)
        atomicAdd(&class_cnt[labels[i]], 1);
    __syncthreads();

    float lsum = 0.0f;
    int   vcnt = 0;
    for (int i = t; i < BATCH_N; i += 256) {
        const int   lb  = labels[i];
        const int   pc  = class_cnt[lb] - 1;
        const float tmp = temps[lb];
        const float pm  = pos_acc[i] / (float)(pc > 0 ? pc : 1);
        const float ld  = logf(denom_acc[i]);     // implicit max = 0 (|logit|<=20)
        const float li  = -(BASE_TEMP / tmp) * (pm - ld);
        if (pc > 0) { lsum += li; vcnt += 1; }
    }
    red_loss[t]  = lsum;
    red_valid[t] = vcnt;
    __syncthreads();

    for (int s = 128; s > 0; s >>= 1) {
        if (t < s) {
            red_loss[t]  += red_loss[t + s];
            red_valid[t] += red_valid[t + s];
        }
        __syncthreads();
    }
    if (t == 0)
        out[0] = red_loss[0] / (float)(red_valid[0] > 0 ? red_valid[0] : 1);
}

// ---------------------------------------------------------------------------
extern "C" void kernel_launch(void* const* d_in, const int* in_sizes, int n_in,
                              void* d_out, int out_size, void* d_ws, size_t ws_size,
                              hipStream_t stream) {
    const float* emb         = (const float*)d_in[0];
    const int*   labels      = (const int*)  d_in[1];
    const float* bank        = (const float*)d_in[2];
    const int*   bank_labels = (const int*)  d_in[3];
    const float* temps       = (const float*)d_in[4];
    float*       out         = (float*)d_out;

    float* denom_acc = (float*)d_ws;              // [2048]
    float* pos_acc   = denom_acc + BATCH_N;       // [2048]

    zero_ws_kernel<<<(2 * BATCH_N + 255) / 256, 256, 0, stream>>>(denom_acc,
                                                                  2 * BATCH_N);

    // 16 row-groups (128 rows) x 36 column chunks (512 cols), 128 threads.
    dim3 grid(BATCH_N / ROWS_PER_WG, NCHUNK);
    supcon_tile_kernel<<<grid, 128, 0, stream>>>(emb, labels, bank, bank_labels,
                                                 temps, denom_acc, pos_acc);

    supcon_finalize_kernel<<<1, 256, 0, stream>>>(labels, temps, denom_acc,
                                                  pos_acc, out);
}